// Sparse_68925635166691
// MI455X (gfx1250) — compile-verified
//
#include <hip/hip_runtime.h>
#include <cmath>

typedef __attribute__((ext_vector_type(2))) float v2f;
typedef __attribute__((ext_vector_type(8))) float v8f;

// Fused: logits = x @ Weff + beff, softmax -> out[B,10]
//   Weff[i,c] = W[i,0]*fcw[c,col0(i)] + W[i,1]*fcw[c,col1(i)]   (784x16, cols 10..15 = 0)
//   beff[c]   = fcb[c] + sum_i b[i,0]*fcw[c,col0(i)] + b[i,1]*fcw[c,col1(i)]
// col map reproduces the reference's buggy wraparound: i=782 -> (0,1); i=783 -> (783,0).
__global__ __launch_bounds__(256) void sparse_fused_kernel(
    const float* __restrict__ x,    // [B,784]
    const float* __restrict__ Ws,   // [784,2]
    const float* __restrict__ bs,   // [784,2]
    const float* __restrict__ fcw,  // [10,784]
    const float* __restrict__ fcb,  // [10]
    float* __restrict__ out,        // [B,10]
    int Bt) {
    constexpr int N = 784;
    __shared__ float sW[16 * N];     // Weff transposed: sW[c*784 + k]  (B-operand source)
    __shared__ float sBias[16];
    __shared__ float sLog[8 * 256];  // per-wave 16x16 logits tile; reused for bias partials

    const int tid = threadIdx.x;

    // ---- 1. build Weff into LDS (padded to 16 classes with zeros) ----
    for (int c = 0; c < 16; ++c) {
        for (int i = tid; i < N; i += 256) {
            float v = 0.0f;
            if (c < 10) {
                const int c0 = (i == 782) ? 0 : i;
                const int c1 = (i == 782) ? 1 : ((i + 1 == N) ? 0 : i + 1);
                v = Ws[2 * i] * fcw[c * N + c0] + Ws[2 * i + 1] * fcw[c * N + c1];
            }
            sW[c * N + i] = v;
        }
    }

    // ---- 2. beff partial sums: thread t -> class (t&15), K-slice (t>>4) of 49 ----
    {
        const int c = tid & 15;
        const int slice = tid >> 4;           // 16 slices * 49 = 784
        float p = 0.0f;
        if (c < 10) {
            const int i0 = slice * 49;
            for (int i = i0; i < i0 + 49; ++i) {
                const int c0 = (i == 782) ? 0 : i;
                const int c1 = (i == 782) ? 1 : ((i + 1 == N) ? 0 : i + 1);
                p += bs[2 * i] * fcw[c * N + c0] + bs[2 * i + 1] * fcw[c * N + c1];
            }
        }
        sLog[slice * 16 + c] = p;
    }
    __syncthreads();
    if (tid < 16) {
        float s = (tid < 10) ? fcb[tid] : 0.0f;
        for (int k = 0; k < 16; ++k) s += sLog[k * 16 + tid];
        sBias[tid] = s;
    }
    __syncthreads();  // sW + sBias ready; sLog free for reuse

    // ---- 3. WMMA GEMM: each wave computes a 16(batch) x 16(class) logits tile ----
    const int wave = tid >> 5;
    const int lane = tid & 31;
    const int col  = lane & 15;   // A: M index / B: N index / D: N index
    const int half = lane >> 4;   // A/B: K pair selector (K = 2*half + {0,1})
    const int rowBase = blockIdx.x * 128 + wave * 16;
    int row = rowBase + col;
    if (row >= Bt) row = Bt - 1;  // clamp keeps EXEC all-ones for WMMA

    const float* xp = x + (long)row * N + 2 * half;   // A frag: x[row, k+2*half .. +1]
    const float* wp = &sW[col * N + 2 * half];        // B frag: Weff[col, k+2*half .. +1]

    v8f acc0 = {};
    v8f acc1 = {};
    for (int kc = 0; kc < N; kc += 16) {
        __builtin_prefetch(xp + kc + 256, 0, 0);      // global_prefetch_b8, ~1KB ahead
#pragma unroll
        for (int u = 0; u < 16; u += 8) {
            v2f a0 = *(const v2f*)(xp + kc + u);
            v2f b0 = *(const v2f*)(wp + kc + u);
            acc0 = __builtin_amdgcn_wmma_f32_16x16x4_f32(
                false, a0, false, b0, (short)0, acc0, false, false);
            v2f a1 = *(const v2f*)(xp + kc + u + 4);
            v2f b1 = *(const v2f*)(wp + kc + u + 4);
            acc1 = __builtin_amdgcn_wmma_f32_16x16x4_f32(
                false, a1, false, b1, (short)0, acc1, false, false);
        }
    }

    // ---- 4. stage tile to LDS (D layout: VGPR r, lane L -> M = r + 8*(L>>4), N = L&15)
    float* myTile = &sLog[wave * 256];
#pragma unroll
    for (int r = 0; r < 8; ++r)
        myTile[(r + 8 * half) * 16 + col] = acc0[r] + acc1[r];
    __syncthreads();

    // ---- 5. fused bias + softmax: lanes 0..15 each own one batch row ----
    if (lane < 16) {
        const int grow = rowBase + lane;
        if (grow < Bt) {
            const float* lrow = &sLog[wave * 256 + lane * 16];
            float vals[10];
            float mx = -1e30f;
#pragma unroll
            for (int c = 0; c < 10; ++c) {
                vals[c] = lrow[c] + sBias[c];
                mx = fmaxf(mx, vals[c]);
            }
            float s = 0.0f;
#pragma unroll
            for (int c = 0; c < 10; ++c) {
                vals[c] = __expf(vals[c] - mx);
                s += vals[c];
            }
            const float inv = 1.0f / s;
            float* orow = out + (long)grow * 10;
#pragma unroll
            for (int c = 0; c < 10; ++c) orow[c] = vals[c] * inv;
        }
    }
}

extern "C" void kernel_launch(void* const* d_in, const int* in_sizes, int n_in,
                              void* d_out, int out_size, void* d_ws, size_t ws_size,
                              hipStream_t stream) {
    const float* x   = (const float*)d_in[0];
    const float* Ws  = (const float*)d_in[1];
    const float* bs  = (const float*)d_in[2];
    const float* fcw = (const float*)d_in[3];
    const float* fcb = (const float*)d_in[4];
    float* out = (float*)d_out;

    const int Bt = in_sizes[0] / 784;          // batch rows
    const int grid = (Bt + 127) / 128;         // 128 rows per block (8 waves x 16)
    sparse_fused_kernel<<<dim3(grid), dim3(256), 0, stream>>>(x, Ws, bs, fcw, fcb, out, Bt);
}